// KokoroModelA_7988639171397
// MI455X (gfx1250) — compile-verified
//
#include <hip/hip_runtime.h>
#include <hip/hip_bf16.h>

// ---------------------------------------------------------------------------
// Kokoro duration->alignment expansion for MI455X (gfx1250, wave32).
//
// aln[b,t,f] is one-hot in t  =>  the einsums are a monotonic gather.
// Workload is HBM-write bound (~110 MB out => ~5.6us @ 23.3 TB/s).
// The matrix op runs on the WMMA path: f32 16x16x4 WMMA with a one-hot B
// fragment synthesized in registers (bit-exact: every accumulated term is
// 0*x or 1*x), skipping all K-blocks outside the tile's token span
// (monotonic alignment => ~1 K-step per tile).
//
// Round-2 fixes (from asm inspection):
//  * kb-loop bounds go through readfirstlane -> scalar uniform loop,
//    EXEC provably all-1s at every v_wmma (ISA requirement), no per-lane
//    loop masking.
//  * A-fragment indices are clamped instead of guarded (B==0 for t>511
//    makes those slots don't-care) -> no s_and_saveexec in the hot loop.
//  * Branchless fixed-9-step lower_bound instead of divergent while-loop.
// ---------------------------------------------------------------------------

#define BATCH  8
#define T_TOK  512
#define NBINS  50
#define MAXF   3000
#define SPF    600.0f
#define CD     640      // en channels   (from d:   [B,T,640])
#define CA     512      // asr channels  (from t_en:[B,512,T])

// Output layout (flat f32 elements, reference return order):
//   asr [B,512,3000] @ 0
//   en  [B,640,3000] @ ASR_N
//   audio_length i32 [B] @ ASR_N+EN_N   (raw int bits)
//   pred_dur f32 [B,512] @ ASR_N+EN_N+B
#define ASR_N  (BATCH * CA * MAXF)
#define EN_N   (BATCH * CD * MAXF)

typedef __attribute__((ext_vector_type(2))) float v2f;
typedef __attribute__((ext_vector_type(8))) float v8f;

// ---------------------------------------------------------------------------
// Kernel 1: per-token durations, block-wide inclusive scan, audio_length.
// One block per batch, 512 threads (= T tokens), 16 waves.
// ---------------------------------------------------------------------------
__global__ void __launch_bounds__(T_TOK)
prep_kernel(const float* __restrict__ dur_logits,
            const int*   __restrict__ amask,
            const float* __restrict__ speed,
            float*       __restrict__ out,
            float*       __restrict__ cum_ws) {
  __shared__ float s[T_TOK];
  const int b = blockIdx.x;
  const int t = threadIdx.x;

  const float* lg = dur_logits + ((size_t)b * T_TOK + t) * NBINS;
  float sum = 0.0f;
  #pragma unroll
  for (int i = 0; i < NBINS; ++i) {
    float x = lg[i];
    sum += 1.0f / (1.0f + expf(-x));          // precise exp: rounding edge safety
  }
  float dur = sum / speed[0];
  float m   = (float)amask[b * T_TOK + t];
  float pd  = fmaxf(rintf(dur), 1.0f) * m;    // rint = round-half-even (RNE mode)

  out[(size_t)ASR_N + EN_N + BATCH + (size_t)b * T_TOK + t] = pd;

  // Hillis-Steele inclusive scan over 512 tokens.
  s[t] = pd;
  __syncthreads();
  for (int off = 1; off < T_TOK; off <<= 1) {
    float v = (t >= off) ? s[t - off] : 0.0f;
    __syncthreads();
    s[t] += v;
    __syncthreads();
  }
  cum_ws[(size_t)b * T_TOK + t] = s[t];

  if (t == T_TOK - 1) {
    // cumsum[-1] is integer-valued; *600 stays < 2^24 => exact f32, then trunc.
    ((int*)out)[(size_t)ASR_N + EN_N + b] = (int)(s[t] * SPF);
  }
}

// ---------------------------------------------------------------------------
// Kernel 2: WMMA expansion.  One wave owns a 16(chan) x 16(frame) C tile.
// grid = (frame_groups=24, batch=8, channel_tile_groups=8), block = 256 (8 waves).
// Channel tiles 0..39 -> en (A from d, coalesced), 40..71 -> asr (A from t_en,
// L2-resident gather).
// ---------------------------------------------------------------------------
__global__ void __launch_bounds__(256)
align_gemm_kernel(const float* __restrict__ d,
                  const float* __restrict__ t_en,
                  const float* __restrict__ cum_ws,
                  float*       __restrict__ out) {
  __shared__ float cum[T_TOK];
  const int fgrp = blockIdx.x;        // 0..23 : 128 frames each
  const int b    = blockIdx.y;        // 0..7
  const int zt   = blockIdx.z;        // 0..7  : 9 channel tiles each

  for (int i = threadIdx.x; i < T_TOK; i += blockDim.x)
    cum[i] = cum_ws[(size_t)b * T_TOK + i];
  __syncthreads();

  const int wave = threadIdx.x >> 5;
  const int lane = threadIdx.x & 31;
  const int f0   = (fgrp * 8 + wave) * 16;
  if (f0 >= MAXF) return;             // whole-wave uniform: EXEC stays all-1s

  const int n    = lane & 15;         // C/D column, B column
  const int half = lane >> 4;         // 0/1
  const int kA   = half * 2;          // K rows this lane carries in A/B frags
  const int f    = f0 + n;

  // Branchless lower_bound: pos = #{t : cumsum[t] <= f} = first t with cum[t] > f.
  // Fixed 9 steps, candidate index pos+s-1 provably <= 510 at every step.
  const float ff    = (float)f;
  const float total = cum[T_TOK - 1];
  int pos = 0;
  #pragma unroll
  for (int s = 256; s >= 1; s >>= 1) {
    int np = pos + s;
    pos = (cum[np - 1] <= ff) ? np : pos;
  }
  const int tf = (f < MAXF && ff < total) ? pos : -1;

  // Wave-uniform token span for this frame tile.
  int tmin = (tf < 0) ? 0x7fffffff : tf;
  int tmax = tf;
  #pragma unroll
  for (int off = 16; off > 0; off >>= 1) {
    tmin = min(tmin, __shfl_xor(tmin, off, 32));
    tmax = max(tmax, __shfl_xor(tmax, off, 32));
  }
  // Force wave-uniform (SGPR) loop bounds: scalar loop, EXEC untouched -> WMMA legal.
  const int kb0 = __builtin_amdgcn_readfirstlane((tmax < 0) ? 4 : (tmin & ~3));
  const int kb1 = __builtin_amdgcn_readfirstlane((tmax < 0) ? 0 : tmax);

  const float* dsrc   = d    + (size_t)b * T_TOK * CD;
  const float* asrc   = t_en + (size_t)b * CA * T_TOK;
  float* en_out  = out + ASR_N + (size_t)b * CD * MAXF;
  float* asr_out = out +          (size_t)b * CA * MAXF;

  for (int cti = 0; cti < 9; ++cti) {
    const int  ct   = zt * 9 + cti;              // 0..71
    const bool isEn = (ct < 40);
    const int  c0   = isEn ? ct * 16 : (ct - 40) * 16;

    v8f acc = {0.f, 0.f, 0.f, 0.f, 0.f, 0.f, 0.f, 0.f};

    for (int kb = kb0; kb <= kb1; kb += 4) {
      const int ta = kb + kA;
      // One-hot B (4x16): lane column n, rows {kA, kA+1}.
      v2f Bf;
      Bf.x = (tf == ta)     ? 1.0f : 0.0f;
      Bf.y = (tf == ta + 1) ? 1.0f : 0.0f;
      // A (16x4): lane row m = n, K rows {kA, kA+1}.  Indices clamped:
      // whenever t > 511 the matching B entry is 0 (tf <= 511), so the
      // loaded value is don't-care -> unconditional loads, no exec masking.
      const int t0c = (ta     < T_TOK) ? ta     : (T_TOK - 1);
      const int t1c = (ta + 1 < T_TOK) ? ta + 1 : (T_TOK - 1);
      v2f A;
      if (isEn) {
        A.x = dsrc[(size_t)t0c * CD + c0 + n];
        A.y = dsrc[(size_t)t1c * CD + c0 + n];
      } else {
        const float* row = asrc + (size_t)(c0 + n) * T_TOK;
        A.x = row[t0c];
        A.y = row[t1c];
      }

#if __has_builtin(__builtin_amdgcn_wmma_f32_16x16x4_f32)
      acc = __builtin_amdgcn_wmma_f32_16x16x4_f32(
          false, A, false, Bf, (short)0, acc, false, false);
#else
      asm volatile("v_wmma_f32_16x16x4_f32 %0, %1, %2, %0"
                   : "+v"(acc) : "v"(A), "v"(Bf));
#endif
    }

    // C/D layout: VGPR r, lane L -> M = r + 8*(L>>4), N = L&15.
    if (f < MAXF) {
      float* obase = isEn ? en_out : asr_out;
      #pragma unroll
      for (int r = 0; r < 8; ++r) {
        const int M = r + half * 8;
        obase[(size_t)(c0 + M) * MAXF + f] = acc[r];
      }
    }
  }
}

// ---------------------------------------------------------------------------
extern "C" void kernel_launch(void* const* d_in, const int* in_sizes, int n_in,
                              void* d_out, int out_size, void* d_ws, size_t ws_size,
                              hipStream_t stream) {
  const float* d      = (const float*)d_in[0];   // [8,512,640]
  const float* t_en   = (const float*)d_in[1];   // [8,512,512]
  const float* dur_lg = (const float*)d_in[2];   // [8,512,50]
  const int*   amask  = (const int*)  d_in[3];   // [8,512]
  const float* speed  = (const float*)d_in[4];   // [1]
  float* out = (float*)d_out;
  float* cum = (float*)d_ws;                     // B*T floats = 16 KB scratch

  prep_kernel<<<dim3(BATCH), dim3(T_TOK), 0, stream>>>(dur_lg, amask, speed, out, cum);
  align_gemm_kernel<<<dim3(24, BATCH, 8), dim3(256), 0, stream>>>(d, t_en, cum, out);
}